// HungarianMatcher_8881992368283
// MI455X (gfx1250) — compile-verified
//
#include <hip/hip_runtime.h>

typedef float v2f __attribute__((ext_vector_type(2)));
typedef float v8f __attribute__((ext_vector_type(8)));

#define B_  64
#define N_  900
#define C_  365
#define M_  64
#define CPAD 368          // C padded to multiple of 4 (K-chunk)
#define LSTR 370          // LDS row stride (even -> 8B-aligned paired loads)
#define NTILES 57         // ceil(900/16)

#define W_CLASS 1.0f
#define W_BBOX  5.0f
#define W_GIOU  2.0f
#define EPS_    1e-6f

// ---------------------------------------------------------------------------
// Kernel 1: fused softmax-stats + one-hot FP32 WMMA class-gather + bbox/GIoU.
// One wave32 per (batch, 16-row tile).
// ---------------------------------------------------------------------------
__global__ __launch_bounds__(32)
void cost_wmma_kernel(const float* __restrict__ logits,
                      const float* __restrict__ pboxes,
                      const int*   __restrict__ labels,
                      const float* __restrict__ tboxes,
                      float*       __restrict__ cost)
{
    __shared__ float slog[16 * LSTR];   // staged logits tile
    __shared__ float sbox[16 * 4];      // pred boxes (cxcywh)
    __shared__ float tbox[M_ * 4];      // tgt boxes (cxcywh)
    __shared__ float smax[16];
    __shared__ float ssum[16];
    __shared__ int   slab[M_];

    const int tile = blockIdx.x;
    const int b    = blockIdx.y;
    const int n0   = tile * 16;
    const int lane = threadIdx.x;

    // ---- stage logits tile (pad cols/rows with -1e30 so exp -> 0 / benign) ----
    const float* lg = logits + ((size_t)b * N_ + n0) * C_;
    for (int idx = lane; idx < 16 * CPAD; idx += 32) {
        int r  = idx / CPAD;
        int cc = idx - r * CPAD;
        float v = -1.0e30f;
        if (cc < C_ && (n0 + r) < N_) v = lg[(size_t)r * C_ + cc];
        slog[r * LSTR + cc] = v;
    }
    // ---- stage boxes / labels ----
    for (int idx = lane; idx < 16 * 4; idx += 32) {
        int r = idx >> 2;
        float v = 0.25f;
        if ((n0 + r) < N_) v = pboxes[((size_t)b * N_ + n0 + r) * 4 + (idx & 3)];
        sbox[idx] = v;
    }
    for (int idx = lane; idx < M_ * 4; idx += 32)
        tbox[idx] = tboxes[(size_t)b * M_ * 4 + idx];
    for (int idx = lane; idx < M_; idx += 32)
        slab[idx] = labels[b * M_ + idx];
    __syncthreads();

    // ---- per-row softmax stats: 2 lanes per row ----
    {
        int r = lane >> 1, part = lane & 1;
        float mx = -3.0e38f;
        for (int cc = part; cc < CPAD; cc += 2)
            mx = fmaxf(mx, slog[r * LSTR + cc]);
        mx = fmaxf(mx, __shfl_xor(mx, 1, 32));
        float se = 0.0f;
        for (int cc = part; cc < CPAD; cc += 2)
            se += __expf(slog[r * LSTR + cc] - mx);
        se += __shfl_xor(se, 1, 32);
        if (part == 0) { smax[r] = mx; ssum[r] = se; }
    }
    __syncthreads();

    // ---- one-hot GEMM: E[16 rows x 64 tgt] via V_WMMA_F32_16X16X4_F32 ----
    // A layout (16x4 f32): lane holds M=lane&15, K = 2*(lane>=16) + vgpr
    // B layout (4x16 f32): lane holds N=lane&15, K = 2*(lane>=16) + vgpr
    const int   lm   = lane & 15;
    const int   hi   = lane >> 4;
    const int   koff = hi * 2;
    const float amax = smax[lm];
    const int   lab0 = slab[lm];
    const int   lab1 = slab[lm + 16];
    const int   lab2 = slab[lm + 32];
    const int   lab3 = slab[lm + 48];

    v8f acc0 = {0.f,0.f,0.f,0.f,0.f,0.f,0.f,0.f};
    v8f acc1 = acc0, acc2 = acc0, acc3 = acc0;

    for (int c = 0; c < CPAD; c += 4) {
        const int k0 = c + koff;
        v2f A;
        A.x = __expf(slog[lm * LSTR + k0]     - amax);
        A.y = __expf(slog[lm * LSTR + k0 + 1] - amax);
        v2f B0, B1, B2, B3;
        B0.x = (lab0 == k0) ? 1.0f : 0.0f;  B0.y = (lab0 == k0 + 1) ? 1.0f : 0.0f;
        B1.x = (lab1 == k0) ? 1.0f : 0.0f;  B1.y = (lab1 == k0 + 1) ? 1.0f : 0.0f;
        B2.x = (lab2 == k0) ? 1.0f : 0.0f;  B2.y = (lab2 == k0 + 1) ? 1.0f : 0.0f;
        B3.x = (lab3 == k0) ? 1.0f : 0.0f;  B3.y = (lab3 == k0 + 1) ? 1.0f : 0.0f;

        acc0 = __builtin_amdgcn_wmma_f32_16x16x4_f32(false, A, false, B0, (short)0, acc0, false, false);
        acc1 = __builtin_amdgcn_wmma_f32_16x16x4_f32(false, A, false, B1, (short)0, acc1, false, false);
        acc2 = __builtin_amdgcn_wmma_f32_16x16x4_f32(false, A, false, B2, (short)0, acc2, false, false);
        acc3 = __builtin_amdgcn_wmma_f32_16x16x4_f32(false, A, false, B3, (short)0, acc3, false, false);
    }

    // ---- epilogue: D layout 16x16 f32: VGPR r -> row = r + 8*hi, col = lm ----
    float* outb = cost + ((size_t)b * N_ + n0) * M_;
#pragma unroll
    for (int r = 0; r < 8; ++r) {
        const int row = r + 8 * hi;
        const int n   = n0 + row;
        if (n >= N_) continue;
        const float inv = 1.0f / ssum[row];
        const float px = sbox[row*4+0], py = sbox[row*4+1];
        const float pw = sbox[row*4+2], ph = sbox[row*4+3];
        const float px0 = px - 0.5f*pw, py0 = py - 0.5f*ph;
        const float px1 = px + 0.5f*pw, py1 = py + 0.5f*ph;
        const float ap  = (px1 - px0) * (py1 - py0);
#pragma unroll
        for (int j = 0; j < 4; ++j) {
            const float ev = (j == 0) ? acc0[r] : (j == 1) ? acc1[r]
                           : (j == 2) ? acc2[r] : acc3[r];
            const int m = j * 16 + lm;
            const float tx = tbox[m*4+0], ty = tbox[m*4+1];
            const float tw = tbox[m*4+2], th = tbox[m*4+3];
            const float tx0 = tx - 0.5f*tw, ty0 = ty - 0.5f*th;
            const float tx1 = tx + 0.5f*tw, ty1 = ty + 0.5f*th;
            const float at  = (tx1 - tx0) * (ty1 - ty0);
            const float ix0 = fmaxf(px0, tx0), iy0 = fmaxf(py0, ty0);
            const float ix1 = fminf(px1, tx1), iy1 = fminf(py1, ty1);
            const float iw  = fmaxf(ix1 - ix0, 0.0f), ih = fmaxf(iy1 - iy0, 0.0f);
            const float inter = iw * ih;
            const float uni   = ap + at - inter;
            const float iou   = inter / (uni + EPS_);
            const float ex0 = fminf(px0, tx0), ey0 = fminf(py0, ty0);
            const float ex1 = fmaxf(px1, tx1), ey1 = fmaxf(py1, ty1);
            const float ae  = (ex1 - ex0) * (ey1 - ey0);
            const float giou = iou - (ae - uni) / (ae + EPS_);
            const float l1 = fabsf(px - tx) + fabsf(py - ty)
                           + fabsf(pw - tw) + fabsf(ph - th);
            const float cls = -(ev * inv);
            outb[(size_t)row * M_ + m] = W_CLASS * cls + W_BBOX * l1 + W_GIOU * (-giou);
        }
    }
}

// ---------------------------------------------------------------------------
// Kernel 2: greedy assignment. One block per batch. Per-row (min, argmin)
// cached in LDS; only rows whose argmin column was just masked re-scan
// their row (L2-resident). Tie-break matches jnp.argmin (row-major first).
// ---------------------------------------------------------------------------
__global__ __launch_bounds__(256)
void greedy_kernel(const float* __restrict__ cost,
                   int* __restrict__ P, int* __restrict__ T)
{
    __shared__ float rowmin[N_];
    __shared__ short rowarg[N_];
    __shared__ unsigned char colmask[M_];
    __shared__ float redv[256];
    __shared__ int   redi[256];

    const int b   = blockIdx.x;
    const int tid = threadIdx.x;
    const float* Cb = cost + (size_t)b * N_ * M_;

    for (int m = tid; m < M_; m += 256) colmask[m] = 0;
    for (int r = tid; r < N_; r += 256) {
        const float* row = Cb + (size_t)r * M_;
        float mv = 3.0e38f; int ma = 0;
        for (int m = 0; m < M_; ++m) {
            float v = row[m];
            if (v < mv) { mv = v; ma = m; }
        }
        rowmin[r] = mv; rowarg[r] = (short)ma;
    }
    __syncthreads();

    for (int k = 0; k < M_; ++k) {
        // lexicographic (value, row) reduction over 900 rows
        float mv = 3.0e38f; int mr = N_;
        for (int r = tid; r < N_; r += 256) {
            float v = rowmin[r];
            if (v < mv) { mv = v; mr = r; }
        }
        redv[tid] = mv; redi[tid] = mr;
        __syncthreads();
        for (int s = 128; s > 0; s >>= 1) {
            if (tid < s) {
                float v2 = redv[tid + s]; int r2 = redi[tid + s];
                if (v2 < redv[tid] || (v2 == redv[tid] && r2 < redi[tid])) {
                    redv[tid] = v2; redi[tid] = r2;
                }
            }
            __syncthreads();
        }
        const int p = redi[0];
        const int t = rowarg[p];
        if (tid == 0) {
            P[b * M_ + k] = p;
            T[b * M_ + k] = t;
            rowmin[p] = 3.0e38f;     // remove row
            colmask[t] = 1;          // remove column
        }
        __syncthreads();
        // re-scan only rows whose cached argmin column was just removed
        for (int r = tid; r < N_; r += 256) {
            if (rowmin[r] < 1.0e37f && rowarg[r] == (short)t) {
                const float* row = Cb + (size_t)r * M_;
                float mv2 = 3.0e38f; int ma2 = 0;
                for (int m = 0; m < M_; ++m) {
                    if (!colmask[m]) {
                        float v = row[m];
                        if (v < mv2) { mv2 = v; ma2 = m; }
                    }
                }
                rowmin[r] = mv2; rowarg[r] = (short)ma2;
            }
        }
        __syncthreads();
    }
}

// ---------------------------------------------------------------------------
extern "C" void kernel_launch(void* const* d_in, const int* in_sizes, int n_in,
                              void* d_out, int out_size, void* d_ws, size_t ws_size,
                              hipStream_t stream) {
    (void)in_sizes; (void)n_in; (void)out_size; (void)d_ws; (void)ws_size;
    const float* logits = (const float*)d_in[0];   // [B,N,C]
    const float* pboxes = (const float*)d_in[1];   // [B,N,4]
    const int*   labels = (const int*)d_in[2];     // [B,M]
    const float* tboxes = (const float*)d_in[3];   // [B,M,4]

    float* cost = (float*)d_out;                                   // [B,N,M]
    int*   P    = (int*)(cost + (size_t)B_ * N_ * M_);             // [B,K]
    int*   T    = P + (size_t)B_ * M_;                             // [B,K]

    dim3 g1(NTILES, B_);
    cost_wmma_kernel<<<g1, 32, 0, stream>>>(logits, pboxes, labels, tboxes, cost);
    greedy_kernel<<<B_, 256, 0, stream>>>(cost, P, T);
}